// ScaledDotProductAttention_8761733284074
// MI455X (gfx1250) — compile-verified
//
#include <hip/hip_runtime.h>

typedef float v2f __attribute__((ext_vector_type(2)));
typedef float v8f __attribute__((ext_vector_type(8)));

#define NB    16
#define LQ    2048
#define LK    2048
#define DH    128
#define TEMP_INV (1.0f/11.313708498984761f)
#define MASK_FILL (-1e32f)

#define LDK   132           // padded LDS stride (floats) for K/V tiles -> conflict-free
#define PST   18            // padded LDS stride for P scratch (distinct banks, 8B aligned)
#define NWAVE 8
#define ROWS_PER_WG 128
#define NKT   (LK/16)

__device__ __forceinline__ float rmax16(float v) {
#pragma unroll
  for (int o = 1; o < 16; o <<= 1) v = fmaxf(v, __shfl_xor(v, o, 32));
  return v;
}
__device__ __forceinline__ float rsum16(float v) {
#pragma unroll
  for (int o = 1; o < 16; o <<= 1) v += __shfl_xor(v, o, 32);
  return v;
}

__global__ __launch_bounds__(256, 1)
void sdpa_fwd_kernel(const float* __restrict__ q, const float* __restrict__ k,
                     const float* __restrict__ v, const int* __restrict__ mask,
                     float* __restrict__ outp, float* __restrict__ pmat) {
  // double-buffered K/V tiles + per-wave P bounce scratch (~43 KB total)
  __shared__ float ldsK[2][16 * LDK];
  __shared__ float ldsV[2][16 * LDK];
  __shared__ float ldsP[NWAVE * 16 * PST];

  const int tid  = threadIdx.x;
  const int wave = tid >> 5;
  const int lane = tid & 31;
  const int l16  = lane & 15;
  const int hi   = lane >> 4;                 // 0 (lanes 0-15) / 1 (lanes 16-31)
  const int b    = blockIdx.y;
  const int q0   = blockIdx.x * ROWS_PER_WG;
  const int qrow = q0 + wave * 16 + l16;      // this lane's A-frag row

  // cooperative-load geometry: 512 float4 per 16x128 tile, 2 per thread
  const int row0 = tid >> 5;           // p4 = tid
  const int col0 = (tid & 31) * 4;
  const int row1 = (tid + 256) >> 5;   // p4 = tid + 256
  const int col1 = ((tid + 256) & 31) * 4;

  // ---- preload Q tile as 32 ready A-fragments (float2 per lane per d-chunk) ----
  const float* qp = q + ((size_t)b * LQ + qrow) * DH + 2 * hi;
  v2f qreg[32];
#pragma unroll
  for (int c = 0; c < 32; ++c) qreg[c] = *(const v2f*)(qp + 4 * c);

  float mrun[8], lrun[8];
#pragma unroll
  for (int r = 0; r < 8; ++r) { mrun[r] = -INFINITY; lrun[r] = 0.0f; }

  const float* kb = k + (size_t)b * LK * DH;
  const float* vb = v + (size_t)b * LK * DH;
  const int*   mb = mask + (size_t)b * LK;

  float4 stK0, stK1, stV0, stV1;      // register staging for next tile

  // ================= Pass A: streaming softmax stats =================
  // prologue: stage + store tile 0 into buffer 0
  stK0 = *(const float4*)(kb + (size_t)row0 * DH + col0);
  stK1 = *(const float4*)(kb + (size_t)row1 * DH + col1);
  *(float4*)(&ldsK[0][row0 * LDK + col0]) = stK0;
  *(float4*)(&ldsK[0][row1 * LDK + col1]) = stK1;

  for (int kt = 0; kt < NKT; ++kt) {
    const int kk0 = kt * 16;
    const int cur = kt & 1;
    // issue global loads for next tile BEFORE the barrier (latency hidden by compute)
    if (kt + 1 < NKT) {
      const float* kn = kb + (size_t)(kk0 + 16) * DH;
      stK0 = *(const float4*)(kn + (size_t)row0 * DH + col0);
      stK1 = *(const float4*)(kn + (size_t)row1 * DH + col1);
    }
    __syncthreads();                          // buf[cur] fully stored & prior readers done

    const bool mcol = (mb[kk0 + l16] != 0);
    v8f acc = {};
#pragma unroll
    for (int c = 0; c < 32; ++c) {            // 32 x v_wmma_f32_16x16x4_f32
      v2f bf = *(const v2f*)(&ldsK[cur][l16 * LDK + 4 * c + 2 * hi]);
      acc = __builtin_amdgcn_wmma_f32_16x16x4_f32(false, qreg[c], false, bf,
                                                  (short)0, acc, false, false);
    }
#pragma unroll
    for (int r = 0; r < 8; ++r) {
      float s    = mcol ? MASK_FILL : acc[r] * TEMP_INV;
      float tm   = rmax16(s);
      float mnew = fmaxf(mrun[r], tm);
      float e    = __expf(s - mnew);
      float ts   = rsum16(e);
      lrun[r] = lrun[r] * __expf(mrun[r] - mnew) + ts;
      mrun[r] = mnew;
    }
    // drain staged registers into the other buffer (read of it finished pre-barrier)
    if (kt + 1 < NKT) {
      const int nxt = cur ^ 1;
      *(float4*)(&ldsK[nxt][row0 * LDK + col0]) = stK0;
      *(float4*)(&ldsK[nxt][row1 * LDK + col1]) = stK1;
    }
  }

  float invl[8];
#pragma unroll
  for (int r = 0; r < 8; ++r) invl[r] = 1.0f / lrun[r];   // l >= 1 always

  // ================= Pass B: p write + out = P @ V =================
  v8f oacc[8] = {};

  // prologue: stage + store tile 0 (K and V) into buffer 0
  stK0 = *(const float4*)(kb + (size_t)row0 * DH + col0);
  stK1 = *(const float4*)(kb + (size_t)row1 * DH + col1);
  stV0 = *(const float4*)(vb + (size_t)row0 * DH + col0);
  stV1 = *(const float4*)(vb + (size_t)row1 * DH + col1);
  *(float4*)(&ldsK[0][row0 * LDK + col0]) = stK0;
  *(float4*)(&ldsK[0][row1 * LDK + col1]) = stK1;
  *(float4*)(&ldsV[0][row0 * LDK + col0]) = stV0;
  *(float4*)(&ldsV[0][row1 * LDK + col1]) = stV1;

  for (int kt = 0; kt < NKT; ++kt) {
    const int kk0 = kt * 16;
    const int cur = kt & 1;
    if (kt + 1 < NKT) {
      const float* kn = kb + (size_t)(kk0 + 16) * DH;
      const float* vn = vb + (size_t)(kk0 + 16) * DH;
      stK0 = *(const float4*)(kn + (size_t)row0 * DH + col0);
      stK1 = *(const float4*)(kn + (size_t)row1 * DH + col1);
      stV0 = *(const float4*)(vn + (size_t)row0 * DH + col0);
      stV1 = *(const float4*)(vn + (size_t)row1 * DH + col1);
    }
    __syncthreads();

    const bool mcol = (mb[kk0 + l16] != 0);
    v8f acc = {};
#pragma unroll
    for (int c = 0; c < 32; ++c) {            // recompute S tile
      v2f bf = *(const v2f*)(&ldsK[cur][l16 * LDK + 4 * c + 2 * hi]);
      acc = __builtin_amdgcn_wmma_f32_16x16x4_f32(false, qreg[c], false, bf,
                                                  (short)0, acc, false, false);
    }

    float* ldsPw = &ldsP[wave * 16 * PST];
#pragma unroll
    for (int r = 0; r < 8; ++r) {             // normalized, masked p
      const int row = r + 8 * hi;
      float pv = mcol ? 0.0f : __expf(acc[r] * TEMP_INV - mrun[r]) * invl[r];
      pmat[((size_t)b * LQ + q0 + wave * 16 + row) * LK + kk0 + l16] = pv;
      ldsPw[row * PST + l16] = pv;            // C-layout -> LDS bounce
    }
    __syncthreads();                          // order ldsP writes before A-frag reads

#pragma unroll
    for (int kk = 0; kk < 4; ++kk) {          // P(16x16) @ V(16x128): 4 K-chunks x 8 N-tiles
      v2f a = *(const v2f*)(&ldsPw[l16 * PST + 4 * kk + 2 * hi]);
#pragma unroll
      for (int dt = 0; dt < 8; ++dt) {
        v2f bf;
        bf.x = ldsV[cur][(4 * kk + 2 * hi)     * LDK + 16 * dt + l16];
        bf.y = ldsV[cur][(4 * kk + 2 * hi + 1) * LDK + 16 * dt + l16];
        oacc[dt] = __builtin_amdgcn_wmma_f32_16x16x4_f32(false, a, false, bf,
                                                         (short)0, oacc[dt], false, false);
      }
    }

    if (kt + 1 < NKT) {
      const int nxt = cur ^ 1;
      *(float4*)(&ldsK[nxt][row0 * LDK + col0]) = stK0;
      *(float4*)(&ldsK[nxt][row1 * LDK + col1]) = stK1;
      *(float4*)(&ldsV[nxt][row0 * LDK + col0]) = stV0;
      *(float4*)(&ldsV[nxt][row1 * LDK + col1]) = stV1;
    }
  }

  // ---- store out tile (16 x 128) ----
#pragma unroll
  for (int dt = 0; dt < 8; ++dt)
#pragma unroll
    for (int r = 0; r < 8; ++r)
      outp[((size_t)b * LQ + q0 + wave * 16 + r + 8 * hi) * DH + dt * 16 + l16] =
          oacc[dt][r];
}

extern "C" void kernel_launch(void* const* d_in, const int* in_sizes, int n_in,
                              void* d_out, int out_size, void* d_ws, size_t ws_size,
                              hipStream_t stream) {
  const float* q    = (const float*)d_in[0];
  const float* k    = (const float*)d_in[1];
  const float* v    = (const float*)d_in[2];
  // d_in[3] = diag_mask (unused by forward)
  const int*   mask = (const int*)d_in[4];

  float* out  = (float*)d_out;                          // [B, Lq, D]
  float* pmat = (float*)d_out + (size_t)NB * LQ * DH;   // [B, Lq, Lk]

  dim3 grid(LQ / ROWS_PER_WG, NB);   // 16 x 16
  dim3 block(256);
  sdpa_fwd_kernel<<<grid, block, 0, stream>>>(q, k, v, mask, out, pmat);
}